// BatchEmbeddingUpdater_38826504356034
// MI455X (gfx1250) — compile-verified
//
#include <hip/hip_runtime.h>
#include <hip/hip_bf16.h>

typedef __attribute__((ext_vector_type(2))) float v2f;
typedef __attribute__((ext_vector_type(8))) float v8f;

#define DIM_NODE 128
#define DIM_HID  256

// Workspace layout (floats), per side (stride WS_SIDE):
//   [0 .. 16383]     WfT1[n][k]  = (W_resize @ W_out[0:256,:])^T      (128x128)
//   [16384 .. 32767] WfT2[n][k]  = (W_nig    @ W_out[256:512,:])^T    (128x128)
//   [32768 .. 32895] bf[n] = b_out + b_resize@W_out_top + b_nig@W_out_bot
#define WS_SIDE 32896

// ---------------------------------------------------------------------------
// Stage A: fuse the two back-to-back linears into one 128x128 weight per input
// (exact: the reference has no nonlinearity between them). Output stored
// TRANSPOSED so stage-B WMMA B-operand loads are contiguous float2s.
// 65536 threads, 256 MACs each.
// ---------------------------------------------------------------------------
__global__ void __launch_bounds__(256)
fuse_weights_kernel(const float* __restrict__ Wr0, const float* __restrict__ Wn0,
                    const float* __restrict__ Wo0,
                    const float* __restrict__ Wr1, const float* __restrict__ Wn1,
                    const float* __restrict__ Wo1,
                    float* __restrict__ ws)
{
    int idx  = blockIdx.x * 256 + threadIdx.x;   // 0 .. 65535
    int side = idx >> 15;
    int mat  = (idx >> 14) & 1;
    int n    = (idx >> 7) & 127;
    int k    = idx & 127;

    const float* In = side ? (mat ? Wn1 : Wr1) : (mat ? Wn0 : Wr0);
    const float* Wo = side ? Wo1 : Wo0;

    const float* inRow = In + (size_t)k * DIM_HID;          // W[k][h], h=0..255
    const float* woCol = Wo + (size_t)mat * DIM_HID * DIM_NODE + n; // W_out[mat*256+h][n]

    float acc = 0.f;
#pragma unroll 8
    for (int h = 0; h < DIM_HID; ++h)
        acc = fmaf(inRow[h], woCol[(size_t)h * DIM_NODE], acc);

    ws[(size_t)side * WS_SIDE + (size_t)mat * 16384 + n * DIM_NODE + k] = acc;
}

__global__ void __launch_bounds__(256)
fuse_bias_kernel(const float* __restrict__ br0, const float* __restrict__ bn0,
                 const float* __restrict__ Wo0, const float* __restrict__ bo0,
                 const float* __restrict__ br1, const float* __restrict__ bn1,
                 const float* __restrict__ Wo1, const float* __restrict__ bo1,
                 float* __restrict__ ws)
{
    int idx  = blockIdx.x * 256 + threadIdx.x;   // 0 .. 255
    int side = idx >> 7;
    int n    = idx & 127;

    const float* br = side ? br1 : br0;
    const float* bn = side ? bn1 : bn0;
    const float* Wo = side ? Wo1 : Wo0;
    const float* bo = side ? bo1 : bo0;

    float acc = bo[n];
#pragma unroll 4
    for (int h = 0; h < DIM_HID; ++h) {
        acc = fmaf(br[h], Wo[(size_t)h * DIM_NODE + n], acc);
        acc = fmaf(bn[h], Wo[(size_t)(DIM_HID + h) * DIM_NODE + n], acc);
    }
    ws[(size_t)side * WS_SIDE + 32768 + n] = acc;
}

// ---------------------------------------------------------------------------
// Stage B: node_emb = gather(prev, ids) @ Wf1 + nig @ Wf2 + bf, scattered into
// the (already copied) output table. One wave per 16x16 output tile using
// V_WMMA_F32_16X16X4_F32; 8 waves per block cover one 16x128 row strip.
//
// Per ISA VGPR layouts (wave32):
//   A 16x4 f32 : lane = (k>=2 ? 16 : 0) + m ; VGPR p holds k = 2*(lane>>4)+p
//   B 4x16 f32 : lane = (k>=2 ? 16 : 0) + n ; VGPR p holds k = 2*(lane>>4)+p
//   C/D 16x16  : N = lane&15 ; VGPR v holds M = v + 8*(lane>>4)
// ---------------------------------------------------------------------------
__global__ void __launch_bounds__(256)
gemm_scatter_kernel(const int*   __restrict__ ids,
                    const float* __restrict__ prev,
                    const float* __restrict__ nig,
                    const float* __restrict__ WfT1,   // [n][k] 128x128
                    const float* __restrict__ WfT2,   // [n][k] 128x128
                    const float* __restrict__ bf,     // [128]
                    float*       __restrict__ out)
{
    const int lane = threadIdx.x & 31;
    const int c    = threadIdx.x >> 5;   // col tile 0..7
    const int r    = blockIdx.x;         // row tile 0..511
    const int m    = lane & 15;
    const int half = lane >> 4;          // 0: k%4 in {0,1}; 1: k%4 in {2,3}

    const int rowIdx = r * 16 + m;
    const long long nodeA = (long long)ids[rowIdx];

    // A operands: row m of gathered-prev / nig, this lane's K phase
    const float* aG = prev + nodeA * DIM_NODE + 2 * half;
    const float* aN = nig + (long long)rowIdx * DIM_NODE + 2 * half;

    // B operands: column n of fused weight (stored transposed -> contiguous k)
    const int n = c * 16 + m;
    const float* b1 = WfT1 + (size_t)n * DIM_NODE + 2 * half;
    const float* b2 = WfT2 + (size_t)n * DIM_NODE + 2 * half;

    // Bias depends only on N -> same value in all 8 accumulator VGPRs
    const float bias = bf[n];
    v8f acc = {bias, bias, bias, bias, bias, bias, bias, bias};

#pragma unroll
    for (int kc = 0; kc < 32; ++kc) {                 // K = 128, 4 per WMMA
        v2f a = *(const v2f*)(aG + 4 * kc);
        v2f b = *(const v2f*)(b1 + 4 * kc);
        acc = __builtin_amdgcn_wmma_f32_16x16x4_f32(
                  false, a, false, b, (short)0, acc, false, false);
    }
#pragma unroll
    for (int kc = 0; kc < 32; ++kc) {
        v2f a = *(const v2f*)(aN + 4 * kc);
        v2f b = *(const v2f*)(b2 + 4 * kc);
        acc = __builtin_amdgcn_wmma_f32_16x16x4_f32(
                  false, a, false, b, (short)0, acc, false, false);
    }

    // Scatter: VGPR v holds element (M = v + 8*half, N = n) of the tile
#pragma unroll
    for (int v = 0; v < 8; ++v) {
        const int M = v + 8 * half;
        const long long node = (long long)ids[r * 16 + M];
        out[node * DIM_NODE + n] = acc[v];
    }
}

extern "C" void kernel_launch(void* const* d_in, const int* in_sizes, int n_in,
                              void* d_out, int out_size, void* d_ws, size_t ws_size,
                              hipStream_t stream)
{
    const int*   src_ids      = (const int*)d_in[0];
    const int*   dst_ids      = (const int*)d_in[1];
    const float* src_prev     = (const float*)d_in[2];
    const float* dst_prev     = (const float*)d_in[3];
    const float* src_nig      = (const float*)d_in[4];
    const float* dst_nig      = (const float*)d_in[5];
    const float* W_src_nig    = (const float*)d_in[6];
    const float* b_src_nig    = (const float*)d_in[7];
    const float* W_src_resize = (const float*)d_in[8];
    const float* b_src_resize = (const float*)d_in[9];
    const float* W_src_out    = (const float*)d_in[10];
    const float* b_src_out    = (const float*)d_in[11];
    const float* W_dst_nig    = (const float*)d_in[12];
    const float* b_dst_nig    = (const float*)d_in[13];
    const float* W_dst_resize = (const float*)d_in[14];
    const float* b_dst_resize = (const float*)d_in[15];
    const float* W_dst_out    = (const float*)d_in[16];
    const float* b_dst_out    = (const float*)d_in[17];

    float* out = (float*)d_out;
    float* ws  = (float*)d_ws;

    const size_t nNodes = (size_t)in_sizes[2] / DIM_NODE;   // 200000
    const int    batch  = in_sizes[0];                      // 8192
    const size_t tableElems = nNodes * DIM_NODE;

    // 1) Copy-through of both tables (the HBM-bound part, ~410 MB total).
    hipMemcpyAsync(out, src_prev, tableElems * sizeof(float),
                   hipMemcpyDeviceToDevice, stream);
    hipMemcpyAsync(out + tableElems, dst_prev, tableElems * sizeof(float),
                   hipMemcpyDeviceToDevice, stream);

    // 2) Fuse the linear chain into one 128x128 weight per input + one bias.
    fuse_weights_kernel<<<256, 256, 0, stream>>>(
        W_src_resize, W_src_nig, W_src_out,
        W_dst_resize, W_dst_nig, W_dst_out, ws);
    fuse_bias_kernel<<<1, 256, 0, stream>>>(
        b_src_resize, b_src_nig, W_src_out, b_src_out,
        b_dst_resize, b_dst_nig, W_dst_out, b_dst_out, ws);

    // 3) WMMA GEMM + scatter for each side (in-order on `stream`, so the
    //    scatter lands after the memcpy of the same region).
    gemm_scatter_kernel<<<batch / 16, 256, 0, stream>>>(
        src_ids, src_prev, src_nig,
        ws, ws + 16384, ws + 32768, out);
    gemm_scatter_kernel<<<batch / 16, 256, 0, stream>>>(
        dst_ids, dst_prev, dst_nig,
        ws + WS_SIDE, ws + WS_SIDE + 16384, ws + WS_SIDE + 32768,
        out + tableElems);
}